// FlashInferAttention_17325898072394
// MI455X (gfx1250) — compile-verified
//
#include <hip/hip_runtime.h>
#include <math.h>

// ---------------- constants ----------------
#define HIDDEN   2048
#define N_HEADS  16
#define N_KV     4
#define HEAD_DIM 128
#define Q_SIZE   (N_HEADS * HEAD_DIM)          // 2048
#define KV_SIZE  (N_KV * HEAD_DIM)             // 512
#define QKV_N    (Q_SIZE + 2 * KV_SIZE)        // 3072
#define B_       32
#define CACHE_LEN 8192
#define NSPLIT   16
#define CHUNK    (CACHE_LEN / NSPLIT)          // 512
#define SCALE    0.08838834764831845f          // 1/sqrt(128)
#define EPSF     1e-6f

typedef __attribute__((ext_vector_type(2))) float v2f;
typedef __attribute__((ext_vector_type(8))) float v8f;

// ---------------------------------------------------------------------------
// Kernel 1/5: fp32 WMMA GEMM  C[M=32,N] = A[32,K] @ B[K,N] (+bias)
// One 16x16 tile per wave, K-loop of V_WMMA_F32_16X16X4_F32.
// A 16x4 layout : lane<16 -> M=lane, v0=K+0,v1=K+1 ; lane>=16 -> K+2,K+3
// B 4x16 layout : v0 = B[K+2*hi][n0+lo], v1 = B[K+1+2*hi][n0+lo]
// C 16x16 layout: VGPR r -> M = r + 8*hi, N = n0+lo
// ---------------------------------------------------------------------------
__global__ void gemm_wmma_kernel(const float* __restrict__ A,
                                 const float* __restrict__ Bm,
                                 const float* __restrict__ bias,
                                 float* __restrict__ C,
                                 int K, int N) {
    const int lane = threadIdx.x & 31;
    const int wave = threadIdx.x >> 5;
    const int tile = blockIdx.x * 8 + wave;
    const int ntn  = N >> 4;
    const int nt   = tile % ntn;
    const int mt   = tile / ntn;
    const int n0 = nt << 4, m0 = mt << 4;
    const int hi = lane >> 4, lo = lane & 15;

    v8f acc = {};
    const float* arow = A + (m0 + lo) * K + 2 * hi;          // A row for this lane
    const float* bcol = Bm + (long)(2 * hi) * N + n0 + lo;   // B col for this lane

    for (int k = 0; k < K; k += 4) {
        v2f a, b;
        a[0] = arow[k];
        a[1] = arow[k + 1];
        b[0] = bcol[(long)k * N];
        b[1] = bcol[(long)(k + 1) * N];
        acc = __builtin_amdgcn_wmma_f32_16x16x4_f32(
                  false, a, false, b, (short)0, acc, false, false);
    }

    const float bv = bias ? bias[n0 + lo] : 0.0f;
#pragma unroll
    for (int r = 0; r < 8; ++r) {
        const int row = m0 + r + 8 * hi;
        C[(long)row * N + n0 + lo] = acc[r] + bv;
    }
}

// ---------------------------------------------------------------------------
// Kernel 2/5: per-head RMSNorm of q,k ; copy v.  grid = B*24, block = 128
// ---------------------------------------------------------------------------
__global__ void qkv_post_kernel(const float* __restrict__ qkv,
                                const float* __restrict__ qg,
                                const float* __restrict__ kg,
                                float* __restrict__ qn,
                                float* __restrict__ kn,
                                float* __restrict__ vn) {
    const int b   = blockIdx.x / 24;
    const int idx = blockIdx.x % 24;
    const int d   = threadIdx.x;
    __shared__ float sred[HEAD_DIM];

    if (idx < 16) {                                   // q heads
        const float* x = qkv + b * QKV_N + idx * HEAD_DIM;
        float v = x[d];
        sred[d] = v * v; __syncthreads();
        for (int st = 64; st > 0; st >>= 1) { if (d < st) sred[d] += sred[d + st]; __syncthreads(); }
        float ms = sred[0] * (1.0f / HEAD_DIM);
        qn[(b * N_HEADS + idx) * HEAD_DIM + d] = v * rsqrtf(ms + EPSF) * qg[d];
    } else if (idx < 20) {                            // k heads
        const int h = idx - 16;
        const float* x = qkv + b * QKV_N + Q_SIZE + h * HEAD_DIM;
        float v = x[d];
        sred[d] = v * v; __syncthreads();
        for (int st = 64; st > 0; st >>= 1) { if (d < st) sred[d] += sred[d + st]; __syncthreads(); }
        float ms = sred[0] * (1.0f / HEAD_DIM);
        kn[(b * N_KV + h) * HEAD_DIM + d] = v * rsqrtf(ms + EPSF) * kg[d];
    } else {                                          // v heads: plain copy
        const int h = idx - 20;
        vn[(b * N_KV + h) * HEAD_DIM + d] =
            qkv[b * QKV_N + Q_SIZE + KV_SIZE + h * HEAD_DIM + d];
    }
}

// ---------------------------------------------------------------------------
// Kernel 3/5: flash-decode split.  grid = (NSPLIT, N_KV, B), block = 256.
// One wave per key row; lane holds dims 4*lane..4*lane+3 as float4.
// Online softmax over the 4 GQA query heads of this KV head.
// ---------------------------------------------------------------------------
__global__ void attn_split_kernel(const float* __restrict__ kc,
                                  const float* __restrict__ vc,
                                  const float* __restrict__ qn,
                                  float* __restrict__ pacc,
                                  float* __restrict__ pms) {
    const int split = blockIdx.x, kv = blockIdx.y, b = blockIdx.z;
    const int tid = threadIdx.x, lane = tid & 31, wave = tid >> 5;

    __shared__ float red_m[8][4], red_s[8][4];
    __shared__ float red_acc[8][4][HEAD_DIM];
    __shared__ float fac[8][4], gmx[4], gsx[4];

    // q registers: 4 heads x 4 dims per lane
    float4 q[4];
#pragma unroll
    for (int h = 0; h < 4; ++h)
        q[h] = *(const float4*)(qn + ((b * N_HEADS + kv * 4 + h) * HEAD_DIM) + 4 * lane);

    float  m[4] = {-1e30f, -1e30f, -1e30f, -1e30f};
    float  s[4] = {0.f, 0.f, 0.f, 0.f};
    float4 a4[4];
#pragma unroll
    for (int h = 0; h < 4; ++h) a4[h] = make_float4(0.f, 0.f, 0.f, 0.f);

    const long rowstride = (long)N_KV * HEAD_DIM;            // floats between key rows
    const int  base = split * CHUNK + wave;

    for (int i = 0; i < CHUNK / 8; ++i) {
        const long key = base + 8 * i;
        const long off = ((long)b * CACHE_LEN + key) * rowstride + (long)kv * HEAD_DIM + 4 * lane;
        // prefetch next iteration's K/V rows (global_prefetch_b8)
        __builtin_prefetch(kc + off + 8 * rowstride, 0, 1);
        __builtin_prefetch(vc + off + 8 * rowstride, 0, 1);

        const float4 kk = *(const float4*)(kc + off);
        float p[4];
#pragma unroll
        for (int h = 0; h < 4; ++h)
            p[h] = q[h].x * kk.x + q[h].y * kk.y + q[h].z * kk.z + q[h].w * kk.w;
        // wave32 butterfly reduce -> all lanes hold full dot
#pragma unroll
        for (int off2 = 16; off2 >= 1; off2 >>= 1) {
#pragma unroll
            for (int h = 0; h < 4; ++h) p[h] += __shfl_xor(p[h], off2, 32);
        }
        const float4 vv = *(const float4*)(vc + off);
#pragma unroll
        for (int h = 0; h < 4; ++h) {
            const float sc = p[h] * SCALE;
            const float nm = fmaxf(m[h], sc);
            const float al = __expf(m[h] - nm);
            const float pe = __expf(sc - nm);
            s[h]  = s[h] * al + pe;
            a4[h].x = a4[h].x * al + pe * vv.x;
            a4[h].y = a4[h].y * al + pe * vv.y;
            a4[h].z = a4[h].z * al + pe * vv.z;
            a4[h].w = a4[h].w * al + pe * vv.w;
            m[h] = nm;
        }
    }

    if (lane == 0)
#pragma unroll
        for (int h = 0; h < 4; ++h) { red_m[wave][h] = m[h]; red_s[wave][h] = s[h]; }
#pragma unroll
    for (int h = 0; h < 4; ++h)
        *(float4*)&red_acc[wave][h][4 * lane] = a4[h];
    __syncthreads();

    if (tid < 4) {
        const int h = tid;
        float gm = -1e30f;
        for (int w = 0; w < 8; ++w) gm = fmaxf(gm, red_m[w][h]);
        float gs = 0.f;
        for (int w = 0; w < 8; ++w) {
            const float f = __expf(red_m[w][h] - gm);
            fac[w][h] = f; gs += f * red_s[w][h];
        }
        gmx[h] = gm; gsx[h] = gs;
        float* pm = pms + (((long)(split * B_ + b) * N_HEADS) + kv * 4 + h) * 2;
        pm[0] = gm; pm[1] = gs;
    }
    __syncthreads();

    for (int pidx = tid; pidx < 4 * HEAD_DIM; pidx += 256) {
        const int h = pidx >> 7, d = pidx & 127;
        float t = 0.f;
#pragma unroll
        for (int w = 0; w < 8; ++w) t += fac[w][h] * red_acc[w][h][d];
        pacc[(((long)(split * B_ + b) * N_HEADS) + kv * 4 + h) * HEAD_DIM + d] = t;
    }
}

// ---------------------------------------------------------------------------
// Kernel 4/5: combine splits + appended-token term.  grid = B*16, block = 128
// ---------------------------------------------------------------------------
__global__ void attn_combine_kernel(const float* __restrict__ qn,
                                    const float* __restrict__ kn,
                                    const float* __restrict__ vn,
                                    const float* __restrict__ pacc,
                                    const float* __restrict__ pms,
                                    float* __restrict__ attn) {
    const int b  = blockIdx.x >> 4;
    const int h  = blockIdx.x & 15;
    const int kv = h >> 2;
    const int d  = threadIdx.x;
    __shared__ float sred[HEAD_DIM];

    // new-token score = (q . k_new) * scale
    const float qv = qn[(b * N_HEADS + h) * HEAD_DIM + d];
    const float kvv = kn[(b * N_KV + kv) * HEAD_DIM + d];
    sred[d] = qv * kvv; __syncthreads();
    for (int st = 64; st > 0; st >>= 1) { if (d < st) sred[d] += sred[d + st]; __syncthreads(); }
    const float sn = sred[0] * SCALE;

    float M = -1e30f;
    for (int sp = 0; sp < NSPLIT; ++sp)
        M = fmaxf(M, pms[((long)(sp * B_ + b) * N_HEADS + h) * 2]);
    const float M2 = fmaxf(M, sn);

    float num = 0.f, den = 0.f;
    for (int sp = 0; sp < NSPLIT; ++sp) {
        const long ix = ((long)(sp * B_ + b) * N_HEADS + h);
        const float f = __expf(pms[ix * 2] - M2);
        num += f * pacc[ix * HEAD_DIM + d];
        den += f * pms[ix * 2 + 1];
    }
    const float fn = __expf(sn - M2);
    num += fn * vn[(b * N_KV + kv) * HEAD_DIM + d];
    den += fn;

    attn[(b * N_HEADS + h) * HEAD_DIM + d] = num / den;
}

// ---------------------------------------------------------------------------
// launch
// ---------------------------------------------------------------------------
extern "C" void kernel_launch(void* const* d_in, const int* in_sizes, int n_in,
                              void* d_out, int out_size, void* d_ws, size_t ws_size,
                              hipStream_t stream) {
    const float* hs     = (const float*)d_in[0];   // [32, 2048]
    const float* kcache = (const float*)d_in[1];   // [32, 8192, 4, 128]
    const float* vcache = (const float*)d_in[2];
    const float* w_qkv  = (const float*)d_in[3];   // [2048, 3072]
    const float* b_qkv  = (const float*)d_in[4];   // [3072]
    const float* w_o    = (const float*)d_in[5];   // [2048, 2048]
    const float* q_g    = (const float*)d_in[6];   // [128]
    const float* k_g    = (const float*)d_in[7];   // [128]
    float* out = (float*)d_out;                    // [32, 2048]

    float* ws = (float*)d_ws;
    float* ws_qkv  = ws;                                   // 32*3072   =  98304
    float* ws_qn   = ws_qkv + B_ * QKV_N;                  // 32*16*128 =  65536
    float* ws_kn   = ws_qn  + B_ * N_HEADS * HEAD_DIM;     // 32*4*128  =  16384
    float* ws_vn   = ws_kn  + B_ * N_KV * HEAD_DIM;        // 32*4*128  =  16384
    float* ws_attn = ws_vn  + B_ * N_KV * HEAD_DIM;        // 32*2048   =  65536
    float* ws_pacc = ws_attn + B_ * Q_SIZE;                // 16*32*16*128 = 1048576
    float* ws_pms  = ws_pacc + (long)NSPLIT * B_ * N_HEADS * HEAD_DIM; // 16*32*16*2

    // 1) QKV projection (WMMA fp32): 384 tiles, 8 waves/block -> 48 blocks
    gemm_wmma_kernel<<<48, 256, 0, stream>>>(hs, w_qkv, b_qkv, ws_qkv, HIDDEN, QKV_N);

    // 2) per-head RMSNorm q/k, copy v
    qkv_post_kernel<<<B_ * 24, HEAD_DIM, 0, stream>>>(ws_qkv, q_g, k_g, ws_qn, ws_kn, ws_vn);

    // 3) flash-decode over the KV cache (bandwidth-dominant stage)
    attn_split_kernel<<<dim3(NSPLIT, N_KV, B_), 256, 0, stream>>>(
        kcache, vcache, ws_qn, ws_pacc, ws_pms);

    // 4) combine splits + new-token K/V
    attn_combine_kernel<<<B_ * N_HEADS, HEAD_DIM, 0, stream>>>(
        ws_qn, ws_kn, ws_vn, ws_pacc, ws_pms, ws_attn);

    // 5) output projection (WMMA fp32): 256 tiles -> 32 blocks
    gemm_wmma_kernel<<<32, 256, 0, stream>>>(ws_attn, w_o, nullptr, out, Q_SIZE, HIDDEN);
}